// GRU_P_31585189494800
// MI455X (gfx1250) — compile-verified
//
#include <hip/hip_runtime.h>

// GRU scan on gfx1250:
//   per wave: 16 batch rows, h kept in registers, w_hh held as WMMA A-operands
//   in registers for the whole T loop; 4x v_wmma_f32_16x16x4_f32 per timestep.
//
// WMMA f32 16x16x4 operand layout assumptions (CDNA5 ISA 7.12.2):
//   A (16x4, MxK):  lane 0-15 -> M=lane ; v0 = K0 | K2 (lane halves), v1 = K1 | K3
//   B (4x16, KxN):  lane 0-15 -> N=lane ; v0 = K0 | K2,               v1 = K1 | K3
//   C/D (16x16):    vgpr r: lanes 0-15 (M=r, N=lane), lanes 16-31 (M=8+r, N=lane-16)
//
// D layout = [gate M][batch N]:
//   RZ tile: lanes-lo hold r-gate pre-acts, lanes-hi hold z-gate pre-acts
//   N  tile: lanes-lo hold n-gate pre-acts, lanes-hi are exact zeros (A rows padded)
// h is stored 4 values/lane: lane-lo = h{0,1,4,5}, lane-hi = h{2,3,6,7}, which IS
// the B-operand layout, so no data movement between timesteps.

typedef float v2f __attribute__((ext_vector_type(2)));
typedef float v8f __attribute__((ext_vector_type(8)));

#define WMMA_F32X4(A, B, C) \
    __builtin_amdgcn_wmma_f32_16x16x4_f32(false, (A), false, (B), (short)0, (C), false, false)

constexpr int Tn = 1024;
constexpr int Bn = 4096;

__device__ __forceinline__ float fast_sigmoid(float v) {
    return __builtin_amdgcn_rcpf(1.0f + __expf(-v));
}
__device__ __forceinline__ float fast_tanh(float v) {
    // tanh(x) = 1 - 2/(e^{2x}+1); saturates correctly at +/-inf
    return 1.0f - 2.0f * __builtin_amdgcn_rcpf(__expf(2.0f * v) + 1.0f);
}

__global__ __launch_bounds__(256) void gru_wmma_kernel(
    const float* __restrict__ x,     // (T, B, 2)
    const float* __restrict__ h0,    // (1, B, 8)
    const float* __restrict__ w_ih,  // (24, 2)
    const float* __restrict__ w_hh,  // (24, 8)
    const float* __restrict__ b_ih,  // (24,)
    const float* __restrict__ b_hh,  // (24,)
    const float* __restrict__ fc_w,  // (1, 8)
    const float* __restrict__ fc_b,  // (1,)
    float* __restrict__ out,         // (T, B, 1)
    float* __restrict__ h_out)       // (1, B, 8)
{
    const int lane  = threadIdx.x & 31;
    const int wid   = (blockIdx.x * blockDim.x + threadIdx.x) >> 5;
    const int m     = lane & 15;           // M-row / batch-column index within tile
    const bool lo   = lane < 16;
    const int bbase = wid * 16;
    const int b     = bbase + m;           // batch element mirrored by this lane

    // ---------------- A operands (w_hh), resident for whole loop ----------------
    const int k0 = lo ? 0 : 2;             // K pair for this lane half
    v2f aRZ1, aRZ2, aN1, aN2;
    aRZ1.x = w_hh[m * 8 + k0 + 0]; aRZ1.y = w_hh[m * 8 + k0 + 1];   // gates 0..15, K 0..3
    aRZ2.x = w_hh[m * 8 + k0 + 4]; aRZ2.y = w_hh[m * 8 + k0 + 5];   // gates 0..15, K 4..7
    if (m < 8) {                                                     // gates 16..23 (n), rest 0
        aN1.x = w_hh[(16 + m) * 8 + k0 + 0]; aN1.y = w_hh[(16 + m) * 8 + k0 + 1];
        aN2.x = w_hh[(16 + m) * 8 + k0 + 4]; aN2.y = w_hh[(16 + m) * 8 + k0 + 5];
    } else {
        aN1.x = aN1.y = aN2.x = aN2.y = 0.0f;
    }

    // ---------------- C bias tiles (b_ih+b_hh for r/z; b_hh only for n) --------
    v8f cRZ, cN;
    #pragma unroll
    for (int r = 0; r < 8; ++r) {
        const int g = lo ? r : 8 + r;
        cRZ[r] = b_ih[g] + b_hh[g];
        cN[r]  = lo ? b_hh[16 + r] : 0.0f;
    }

    // ---------------- per-lane input-path weights ------------------------------
    float wA0[8], wA1[8];                  // lane-lo: r rows, lane-hi: z rows
    #pragma unroll
    for (int j = 0; j < 8; ++j) {
        const int g = lo ? j : 8 + j;
        wA0[j] = w_ih[g * 2 + 0];
        wA1[j] = w_ih[g * 2 + 1];
    }
    const int jl[4] = {0, 1, 4, 5};
    const int jh[4] = {2, 3, 6, 7};
    float wN0[4], wN1[4], bN[4], fcw[4];
    #pragma unroll
    for (int s = 0; s < 4; ++s) {
        const int j = lo ? jl[s] : jh[s];  // hidden unit owned by slot s on this half
        wN0[s] = w_ih[(16 + j) * 2 + 0];
        wN1[s] = w_ih[(16 + j) * 2 + 1];
        bN[s]  = b_ih[16 + j];
        fcw[s] = fc_w[j];
    }
    const float fcb = fc_b[0];

    // ---------------- initial hidden state, slot layout ------------------------
    float h[4];
    {
        const float* hp = h0 + (size_t)b * 8 + (lo ? 0 : 2);
        h[0] = hp[0]; h[1] = hp[1]; h[2] = hp[4]; h[3] = hp[5];
    }

    const float* xp   = x + (size_t)b * 2;
    float*       outp = out + bbase + m;

    for (int t = 0; t < Tn; ++t) {
        const float x0 = xp[0];
        const float x1 = xp[1];
        xp += (size_t)Bn * 2;

        // B operand = h in its resident layout
        v2f b1, b2;
        b1.x = h[0]; b1.y = h[1];
        b2.x = h[2]; b2.y = h[3];

        v8f dRZ = WMMA_F32X4(aRZ1, b1, cRZ);
        dRZ     = WMMA_F32X4(aRZ2, b2, dRZ);
        v8f dN  = WMMA_F32X4(aN1, b1, cN);
        dN      = WMMA_F32X4(aN2, b2, dN);

        // a[j]: lanes-lo = r_j, lanes-hi = z_j ; o[j] = other half's value
        float a[8], o[8], g2[8];
        #pragma unroll
        for (int j = 0; j < 8; ++j) {
            const float u = dRZ[j] + fmaf(x1, wA1[j], x0 * wA0[j]);
            a[j]  = fast_sigmoid(u);
            o[j]  = __shfl_xor(a[j], 16, 32);
            g2[j] = __shfl_xor(dN[j], 16, 32);   // hn for lanes-hi (their dN is 0)
        }

        #pragma unroll
        for (int s = 0; s < 4; ++s) {
            const float rS  = lo ? a[jl[s]]  : o[jh[s]];
            const float zS  = lo ? o[jl[s]]  : a[jh[s]];
            const float hnS = lo ? dN[jl[s]] : g2[jh[s]];
            const float xnS = fmaf(x1, wN1[s], fmaf(x0, wN0[s], bN[s]));
            const float nS  = fast_tanh(fmaf(rS, hnS, xnS));
            h[s] = nS + zS * (h[s] - nS);        // (1-z)*n + z*h
        }

        // FC head: half-sum + cross-half shuffle reduction
        float acc = fmaf(fcw[3], h[3], fmaf(fcw[2], h[2], fmaf(fcw[1], h[1], fcw[0] * h[0])));
        const float y = fast_sigmoid(acc + __shfl_xor(acc, 16, 32) + fcb);
        if (lo) *outp = y;
        outp += Bn;
    }

    // ---------------- final hidden state ---------------------------------------
    {
        float* hp = h_out + (size_t)b * 8 + (lo ? 0 : 2);
        hp[0] = h[0]; hp[1] = h[1]; hp[4] = h[2]; hp[5] = h[3];
    }
}

extern "C" void kernel_launch(void* const* d_in, const int* in_sizes, int n_in,
                              void* d_out, int out_size, void* d_ws, size_t ws_size,
                              hipStream_t stream) {
    const float* x    = (const float*)d_in[0];
    const float* h0   = (const float*)d_in[1];
    const float* w_ih = (const float*)d_in[2];
    const float* w_hh = (const float*)d_in[3];
    const float* b_ih = (const float*)d_in[4];
    const float* b_hh = (const float*)d_in[5];
    const float* fc_w = (const float*)d_in[6];
    const float* fc_b = (const float*)d_in[7];

    float* out    = (float*)d_out;                 // (T, B, 1) flat
    float* h_last = out + (size_t)Tn * Bn;         // then (1, B, 8) flat

    // 4096 batch rows / 16 per wave = 256 waves = 32 blocks of 8 waves
    const int threads = 256;
    const int blocks  = Bn / 16 / (threads / 32);
    gru_wmma_kernel<<<blocks, threads, 0, stream>>>(
        x, h0, w_ih, w_hh, b_ih, b_hh, fc_w, fc_b, out, h_last);
}